// SingleChannelVNTSimpleWrapper_56212531970672
// MI455X (gfx1250) — compile-verified
//
#include <hip/hip_runtime.h>

// ---------------------------------------------------------------------------
// CDNA5 / gfx1250 implementation of the VNT point-cloud network.
// fp32 WMMA (V_WMMA_F32_16X16X4_F32) for the channel-mixing GEMM stages.
// ---------------------------------------------------------------------------

typedef __attribute__((ext_vector_type(2))) float v2f;
typedef __attribute__((ext_vector_type(8))) float v8f;

#define EPSV 1e-6f
#define KNN 20

__device__ __forceinline__ v8f wmma_f32_16x16x4(v2f a, v2f b, v8f c) {
  // 8-arg form: (neg_a, A, neg_b, B, c_mod, C, reuse_a, reuse_b)
  return __builtin_amdgcn_wmma_f32_16x16x4_f32(false, a, false, b, (short)0, c,
                                               false, false);
}

// ---------------------------------------------------------------- utilities
__global__ void zero_kernel(float* __restrict__ p, int n) {
  int i = blockIdx.x * blockDim.x + threadIdx.x;
  if (i < n) p[i] = 0.f;
}

// Wn[r,:] = W[r,:] / sum(W[r,:])
__global__ void rownorm_kernel(const float* __restrict__ W, float* __restrict__ Wn,
                               int rows, int cols) {
  int r = blockIdx.x * blockDim.x + threadIdx.x;
  if (r >= rows) return;
  float s = 0.f;
  for (int c = 0; c < cols; ++c) s += W[(size_t)r * cols + c];
  float inv = 1.f / s;
  for (int c = 0; c < cols; ++c) Wn[(size_t)r * cols + c] = W[(size_t)r * cols + c] * inv;
}

// ---------------------------------------------------------------- kNN (top-20)
// pairwise[n,m] = 2<xn,xm> - |xn|^2 - |xm|^2 ; keep 20 largest (desc, low idx
// first on ties -> strict '>' while scanning m ascending reproduces top_k).
__global__ void __launch_bounds__(256) knn_kernel(const float* __restrict__ pc,
                                                  int* __restrict__ idx, int N) {
  const int b = blockIdx.y;
  const int n = blockIdx.x * 256 + threadIdx.x;
  const float* pcb = pc + (size_t)b * N * 3;
  const float cx = pcb[n * 3 + 0], cy = pcb[n * 3 + 1], cz = pcb[n * 3 + 2];
  const float xxn = cx * cx + cy * cy + cz * cz;

  float val[KNN];
  int vid[KNN];
#pragma unroll
  for (int j = 0; j < KNN; ++j) { val[j] = -3.4e38f; vid[j] = 0; }

  __shared__ float sx[256], sy[256], sz[256];
  for (int m0 = 0; m0 < N; m0 += 256) {
    const int t = threadIdx.x;
    sx[t] = pcb[(m0 + t) * 3 + 0];
    sy[t] = pcb[(m0 + t) * 3 + 1];
    sz[t] = pcb[(m0 + t) * 3 + 2];
    __syncthreads();
    for (int j = 0; j < 256; ++j) {
      const float mx = sx[j], my = sy[j], mz = sz[j];
      const float d = 2.f * (cx * mx + cy * my + cz * mz) - xxn
                      - (mx * mx + my * my + mz * mz);
      if (d > val[KNN - 1]) {
        const int m = m0 + j;
        // branchless sorted insertion, all reads see pre-insertion values
#pragma unroll
        for (int q = KNN - 1; q > 0; --q) {
          const bool p1 = d > val[q - 1];
          const bool p2 = d > val[q];
          const float nv = p1 ? val[q - 1] : (p2 ? d : val[q]);
          const int ni = p1 ? vid[q - 1] : (p2 ? m : vid[q]);
          val[q] = nv; vid[q] = ni;
        }
        if (d > val[0]) { val[0] = d; vid[0] = m; }
      }
    }
    __syncthreads();
  }
#pragma unroll
  for (int j = 0; j < KNN; ++j) idx[((size_t)b * N + n) * KNN + j] = vid[j];
}

// ------------------------------------------------- stage 1: edge conv (K=3)
// X1[b,o,v,n] = mean_k VNlrelu( Wn_pos @ feat, D_pos @ feat ),
// feat channels = (nbr-ctr, ctr, nbr x ctr).
__global__ void __launch_bounds__(256) edge_kernel(
    const float* __restrict__ pc, const int* __restrict__ idx,
    const float* __restrict__ WnPos, const float* __restrict__ DPos,
    float* __restrict__ X1, int N) {
  const int b = blockIdx.y;
  const int tid = threadIdx.x;
  const int o = tid & 63;    // output channel 0..63
  const int lp = tid >> 6;   // local point 0..3
  const int n = blockIdx.x * 4 + lp;

  __shared__ float sc[4][3];
  __shared__ float sn[4][KNN][3];
  __shared__ float sW[64 * 3], sD[64 * 3];
  const float* pcb = pc + (size_t)b * N * 3;

  if (tid < 4 * KNN) {
    const int l = tid / KNN, k = tid % KNN;
    const int nn = blockIdx.x * 4 + l;
    const int m = idx[((size_t)b * N + nn) * KNN + k];
    sn[l][k][0] = pcb[m * 3 + 0];
    sn[l][k][1] = pcb[m * 3 + 1];
    sn[l][k][2] = pcb[m * 3 + 2];
  } else if (tid < 4 * KNN + 4) {
    const int l = tid - 4 * KNN;
    const int nn = blockIdx.x * 4 + l;
    sc[l][0] = pcb[nn * 3 + 0];
    sc[l][1] = pcb[nn * 3 + 1];
    sc[l][2] = pcb[nn * 3 + 2];
  }
  for (int u = tid; u < 192; u += 256) { sW[u] = WnPos[u]; sD[u] = DPos[u]; }
  __syncthreads();

  const float w0 = sW[o * 3 + 0], w1 = sW[o * 3 + 1], w2 = sW[o * 3 + 2];
  const float g0 = sD[o * 3 + 0], g1 = sD[o * 3 + 1], g2 = sD[o * 3 + 2];
  const float cx = sc[lp][0], cy = sc[lp][1], cz = sc[lp][2];

  float ax = 0.f, ay = 0.f, az = 0.f;
#pragma unroll 4
  for (int k = 0; k < KNN; ++k) {
    const float nx = sn[lp][k][0], ny = sn[lp][k][1], nz = sn[lp][k][2];
    const float e0x = nx - cx, e0y = ny - cy, e0z = nz - cz;
    const float e2x = ny * cz - nz * cy;          // cross(nbr, ctr)
    const float e2y = nz * cx - nx * cz;
    const float e2z = nx * cy - ny * cx;
    const float px = w0 * e0x + w1 * cx + w2 * e2x;
    const float py = w0 * e0y + w1 * cy + w2 * e2y;
    const float pz = w0 * e0z + w1 * cz + w2 * e2z;
    const float qx = g0 * e0x + g1 * cx + g2 * e2x;
    const float qy = g0 * e0y + g1 * cy + g2 * e2y;
    const float qz = g0 * e0z + g1 * cz + g2 * e2z;
    const float dot = px * qx + py * qy + pz * qz;
    const float dns = qx * qx + qy * qy + qz * qz;
    const float s = (dot >= 0.f) ? 0.f : dot / (dns + EPSV);
    ax += px - s * qx; ay += py - s * qy; az += pz - s * qz;
  }
  const float inv = 1.f / (float)KNN;
  const size_t base = (size_t)(b * 64 + o) * 3 * N + n;
  X1[base] = ax * inv;
  X1[base + N] = ay * inv;
  X1[base + 2 * N] = az * inv;
}

// --------------------------------------- stages 2/3/4: WMMA GEMM + VN-lrelu
// out[b,o,v,n] = lrelu( Wn[o,:Cin]@X[b,:,v,n] + biasP[b,o,v],
//                       Dw[o,:Cin]@X[b,:,v,n] + biasD[b,o,v] )
// Also atomically accumulates sum_n(out) into meanAcc[b,o,v].
// Block: 256 threads = 8 waves; wave w -> rows [16w,16w+16); cols = one
// 16-wide n-tile, all three v's kept in-register (needed for the lrelu dot).
__global__ void __launch_bounds__(256) vnt_gemm_lrelu(
    const float* __restrict__ X, const float* __restrict__ Wn,
    const float* __restrict__ Dw, const float* __restrict__ biasP,
    const float* __restrict__ biasD, float* __restrict__ Xout,
    float* __restrict__ meanAcc, int Cin, int ldA, int N) {
  const int b = blockIdx.y;
  const int n0 = blockIdx.x * 16;
  const int tid = threadIdx.x;
  const int lane = tid & 31;
  const int wv = tid >> 5;
  const int o0 = wv * 16;

  // LDS B-tile: Bs[i][v][c] = X[b, i, v, n0+c],  i < Cin (<=128) -> <=24 KB
  __shared__ float Bs[128 * 48];
  const int total = Cin * 48;
  for (int t = tid; t < total; t += 256) {
    const int i = t / 48;
    const int rem = t - i * 48;
    const int v = rem >> 4;
    const int c = rem & 15;
    Bs[t] = X[((size_t)(b * Cin + i) * 3 + v) * N + n0 + c];
  }
  __syncthreads();

  // A (16x4 f32) layout: lanes 0-15 -> K=k0,k0+1 ; lanes 16-31 -> K=k0+2,k0+3
  const int arow = o0 + (lane & 15);
  const int koff = (lane >> 4) << 1;  // 0 or 2
  const int bn = lane & 15;

  const v8f vzero = {0.f, 0.f, 0.f, 0.f, 0.f, 0.f, 0.f, 0.f};
  v8f accP[3], accD[3];
#pragma unroll
  for (int v = 0; v < 3; ++v) { accP[v] = vzero; accD[v] = vzero; }

  for (int k0 = 0; k0 < Cin; k0 += 4) {
    v2f aW, aD;
    const float* wp = Wn + (size_t)arow * ldA + k0 + koff;
    const float* dp = Dw + (size_t)arow * ldA + k0 + koff;
    aW.x = wp[0]; aW.y = wp[1];
    aD.x = dp[0]; aD.y = dp[1];
#pragma unroll
    for (int v = 0; v < 3; ++v) {
      v2f bf;
      const float* bp = Bs + (k0 + koff) * 48 + v * 16 + bn;
      bf.x = bp[0];
      bf.y = bp[48];
      accP[v] = wmma_f32_16x16x4(aW, bf, accP[v]);
      accD[v] = wmma_f32_16x16x4(aD, bf, accD[v]);
    }
  }

  // C/D layout: VGPR r -> row o0 + r + 8*(lane>=16), col n0 + (lane&15)
  const int hi = lane >> 4;
  const int c = lane & 15;
#pragma unroll
  for (int r = 0; r < 8; ++r) {
    const int o = o0 + r + (hi << 3);
    float p0 = accP[0][r], p1 = accP[1][r], p2 = accP[2][r];
    float q0 = accD[0][r], q1 = accD[1][r], q2 = accD[2][r];
    if (biasP) {
      const size_t bb = (size_t)(b * 128 + o) * 3;
      p0 += biasP[bb + 0]; p1 += biasP[bb + 1]; p2 += biasP[bb + 2];
      q0 += biasD[bb + 0]; q1 += biasD[bb + 1]; q2 += biasD[bb + 2];
    }
    const float dot = p0 * q0 + p1 * q1 + p2 * q2;
    const float dns = q0 * q0 + q1 * q1 + q2 * q2;
    const float s = (dot >= 0.f) ? 0.f : dot / (dns + EPSV);
    const float y0 = p0 - s * q0, y1 = p1 - s * q1, y2 = p2 - s * q2;
    if (Xout) {
      const size_t base = (size_t)(b * 128 + o) * 3 * N + n0 + c;
      Xout[base] = y0; Xout[base + N] = y1; Xout[base + 2 * N] = y2;
    }
    float s0 = y0, s1 = y1, s2 = y2;
#pragma unroll
    for (int off = 8; off >= 1; off >>= 1) {
      s0 += __shfl_xor(s0, off, 16);
      s1 += __shfl_xor(s1, off, 16);
      s2 += __shfl_xor(s2, off, 16);
    }
    if (c == 0) {
      const size_t mb = (size_t)(b * 128 + o) * 3;
      atomicAdd(&meanAcc[mb + 0], s0);
      atomicAdd(&meanAcc[mb + 1], s1);
      atomicAdd(&meanAcc[mb + 2], s2);
    }
  }
}

// ------- bias for the concat([net, mean]) halves of the 256-wide weights:
// biasP[b,o,v] = sum_i Wfull[o,128+i] * meanPrev[b,i,v]/N  (same for D)
__global__ void bias_kernel(const float* __restrict__ meanAcc,
                            const float* __restrict__ Wfull,
                            const float* __restrict__ Dfull,
                            float* __restrict__ biasP, float* __restrict__ biasD,
                            int N) {
  const int t = blockIdx.x * blockDim.x + threadIdx.x;
  if (t >= 8 * 128 * 3) return;
  const int v = t % 3;
  const int o = (t / 3) & 127;
  const int b = t / 384;
  const float invN = 1.f / (float)N;
  float sp = 0.f, sd = 0.f;
  for (int i = 0; i < 128; ++i) {
    const float m = meanAcc[(size_t)(b * 128 + i) * 3 + v] * invN;
    sp += Wfull[(size_t)o * 256 + 128 + i] * m;
    sd += Dfull[(size_t)o * 256 + 128 + i] * m;
  }
  biasP[t] = sp;
  biasD[t] = sd;
}

__global__ void finalize_kernel(const float* __restrict__ mean4,
                                float* __restrict__ out, int N) {
  const int t = blockIdx.x * blockDim.x + threadIdx.x;
  if (t >= 8 * 128 * 3) return;
  out[t] = mean4[t] * (1.f / (float)N);  // out[b, o*3+v]
}

// ---------------------------------------------------------------------------
extern "C" void kernel_launch(void* const* d_in, const int* in_sizes, int n_in,
                              void* d_out, int out_size, void* d_ws, size_t ws_size,
                              hipStream_t stream) {
  const int B = 8, N = 2048;
  const float* pc   = (const float*)d_in[0];
  const float* Wpos = (const float*)d_in[1];
  const float* Dpos = (const float*)d_in[2];
  const float* W1 = (const float*)d_in[3];
  const float* D1 = (const float*)d_in[4];
  const float* W2 = (const float*)d_in[5];
  const float* D2 = (const float*)d_in[6];
  const float* W3 = (const float*)d_in[7];
  const float* D3 = (const float*)d_in[8];

  char* ws = (char*)d_ws;
  size_t off = 0;
  auto alloc = [&](size_t bytes) -> void* {
    void* p = ws + off;
    off += (bytes + 255) & ~(size_t)255;
    return p;
  };

  float* WnPos = (float*)alloc(64 * 3 * sizeof(float));
  float* Wn1   = (float*)alloc(128 * 64 * sizeof(float));
  float* Wn2   = (float*)alloc(128 * 256 * sizeof(float));
  float* Wn3   = (float*)alloc(128 * 256 * sizeof(float));
  float* X1    = (float*)alloc((size_t)B * 64 * 3 * N * sizeof(float));
  float* X2    = (float*)alloc((size_t)B * 128 * 3 * N * sizeof(float));
  float* X3    = (float*)alloc((size_t)B * 128 * 3 * N * sizeof(float));
  float* means = (float*)alloc(3 * 3072 * sizeof(float));  // mean2|mean3|mean4
  float* mean2 = means;
  float* mean3 = means + 3072;
  float* mean4 = means + 6144;
  float* biasP3 = (float*)alloc(3072 * sizeof(float));
  float* biasD3 = (float*)alloc(3072 * sizeof(float));
  float* biasP4 = (float*)alloc(3072 * sizeof(float));
  float* biasD4 = (float*)alloc(3072 * sizeof(float));
  int* knnIdx   = (int*)alloc((size_t)B * N * KNN * sizeof(int));
  (void)ws_size; (void)in_sizes; (void)n_in; (void)out_size;

  // 0) zero the mean accumulators (re-zeroed every call: graph-replay safe)
  zero_kernel<<<(3 * 3072 + 255) / 256, 256, 0, stream>>>(means, 3 * 3072);

  // 1) row-normalize the W matrices
  rownorm_kernel<<<1, 64, 0, stream>>>(Wpos, WnPos, 64, 3);
  rownorm_kernel<<<2, 64, 0, stream>>>(W1, Wn1, 128, 64);
  rownorm_kernel<<<2, 64, 0, stream>>>(W2, Wn2, 128, 256);
  rownorm_kernel<<<2, 64, 0, stream>>>(W3, Wn3, 128, 256);

  // 2) kNN
  knn_kernel<<<dim3(N / 256, B), 256, 0, stream>>>(pc, knnIdx, N);

  // 3) stage 1: edge conv + mean over k -> X1 [B,64,3,N]
  edge_kernel<<<dim3(N / 4, B), 256, 0, stream>>>(pc, knnIdx, WnPos, Dpos, X1, N);

  // 4) stage 2 GEMM (K=64) -> X2, mean2
  vnt_gemm_lrelu<<<dim3(N / 16, B), 256, 0, stream>>>(
      X1, Wn1, D1, nullptr, nullptr, X2, mean2, 64, 64, N);

  // 5) concat-bias for stage 3, then stage 3 GEMM (K=128) -> X3, mean3
  bias_kernel<<<(3072 + 255) / 256, 256, 0, stream>>>(mean2, Wn2, D2, biasP3, biasD3, N);
  vnt_gemm_lrelu<<<dim3(N / 16, B), 256, 0, stream>>>(
      X2, Wn2, D2, biasP3, biasD3, X3, mean3, 128, 256, N);

  // 6) concat-bias for stage 4, then stage 4 GEMM (K=128) -> mean4 only
  bias_kernel<<<(3072 + 255) / 256, 256, 0, stream>>>(mean3, Wn3, D3, biasP4, biasD4, N);
  vnt_gemm_lrelu<<<dim3(N / 16, B), 256, 0, stream>>>(
      X3, Wn3, D3, biasP4, biasD4, nullptr, mean4, 128, 256, N);

  // 7) final mean over n -> d_out [8, 384]
  finalize_kernel<<<(3072 + 255) / 256, 256, 0, stream>>>(mean4, (float*)d_out, N);
}